// GELU144_39857296507258
// MI455X (gfx1250) — compile-verified
//
#include <hip/hip_runtime.h>
#include <cstdint>

// ---------------------------------------------------------------------------
// Surprise-gated GELU for MI455X (gfx1250, wave32).
//
// Roofline: 8*4096*1024 f32 in + out = 268 MB  =>  ~11.5 us at 23.3 TB/s.
// Purely HBM-bound; no matmul structure, so no WMMA. The CDNA5-specific path
// used here is the async global->LDS DMA (GLOBAL_LOAD_ASYNC_TO_LDS_B128,
// tracked by ASYNCcnt, drained with s_wait_asynccnt) to stage each row once
// in LDS and reuse it for both the z-score pass and the GELU output pass.
//
// Top-k: one wave owns one row. |z| >= 0 so float order == unsigned order of
// the bit patterns; exact binary search on the 31-bit pattern space, counting
// #{z_bits >= mid} per iteration with ballot_w32 + popcount (wave-uniform
// scalar count, no LDS, no barriers). Ties at the K-th value are handled via
// sum_top = sum(z > T) + (K - cnt_gt) * T, which matches a sorted top-k.
// ---------------------------------------------------------------------------

#define D_DIM           1024
#define TOPK            32
#define BLOCK           256
#define WAVES_PER_BLOCK (BLOCK / 32)
#define ELEMS_PER_LANE  (D_DIM / 32)      // 32 elements per lane
#define CHUNKS          (ELEMS_PER_LANE / 4)  // 8 x float4 per lane

__global__ __launch_bounds__(BLOCK) void surprise_gelu_topk_kernel(
    const float* __restrict__ x,
    const float* __restrict__ log_alpha,
    const float* __restrict__ log_sigma,
    const float* __restrict__ ema_mean,
    const float* __restrict__ ema_sq,
    float* __restrict__ out,
    int n_rows)
{
  __shared__ __align__(16) float tile[WAVES_PER_BLOCK * D_DIM];  // 32 KB

  const int lane = threadIdx.x & 31;
  const int wid  = threadIdx.x >> 5;
  const int row  = blockIdx.x * WAVES_PER_BLOCK + wid;
  if (row >= n_rows) return;   // uniform per wave; no barriers anywhere below

  const float* __restrict__ xrow = x   + (size_t)row * D_DIM;
  float* __restrict__       orow = out + (size_t)row * D_DIM;

  // Low 32 bits of a generic pointer to LDS == addrspace(3) byte offset.
  const uint32_t ldsBase = (uint32_t)(uintptr_t)(&tile[wid * D_DIM]);
  const float* __restrict__ ltile = &tile[wid * D_DIM];

  // ---- CDNA5 async DMA: stage the whole row global -> LDS (ASYNCcnt) ----
  #pragma unroll
  for (int c = 0; c < CHUNKS; ++c) {
    const int e = c * 128 + lane * 4;                       // element index
    uint64_t ga = (uint64_t)(uintptr_t)(xrow + e);          // 16B-aligned
    uint32_t la = ldsBase + (uint32_t)(e * 4);              // LDS byte offset
    asm volatile("global_load_async_to_lds_b128 %0, %1, off"
                 :: "v"(la), "v"(ga) : "memory");
  }
  asm volatile("s_wait_asynccnt 0" ::: "memory");
  // Producer and consumer are the same wave -> no workgroup barrier needed.

  // ---- z = |x - mean| * rsqrt(clip(sq - mean^2, 1e-6)), keep bit patterns --
  uint32_t zb[ELEMS_PER_LANE];
  #pragma unroll
  for (int c = 0; c < CHUNKS; ++c) {
    const int e = c * 128 + lane * 4;
    const float4 xv = *(const float4*)(ltile + e);
    const float4 mv = *(const float4*)(ema_mean + e);       // 4 KB, L2-hot
    const float4 qv = *(const float4*)(ema_sq + e);
    zb[c*4+0] = __float_as_uint(fabsf(xv.x - mv.x) * rsqrtf(fmaxf(qv.x - mv.x*mv.x, 1e-6f)));
    zb[c*4+1] = __float_as_uint(fabsf(xv.y - mv.y) * rsqrtf(fmaxf(qv.y - mv.y*mv.y, 1e-6f)));
    zb[c*4+2] = __float_as_uint(fabsf(xv.z - mv.z) * rsqrtf(fmaxf(qv.z - mv.z*mv.z, 1e-6f)));
    zb[c*4+3] = __float_as_uint(fabsf(xv.w - mv.w) * rsqrtf(fmaxf(qv.w - mv.w*mv.w, 1e-6f)));
  }

  // ---- exact top-K threshold: binary search on the unsigned bit pattern ----
  // Invariant: count(z_bits >= lo) >= K  and  count(z_bits >= hi) < K.
  uint32_t lo = 0u, hi = 0x7F800000u;   // z is finite and non-negative
  while (hi - lo > 1u) {
    const uint32_t mid = lo + ((hi - lo) >> 1);
    int cnt = 0;
    #pragma unroll
    for (int j = 0; j < ELEMS_PER_LANE; ++j)
      cnt += __builtin_popcount(__builtin_amdgcn_ballot_w32(zb[j] >= mid));
    if (cnt >= TOPK) lo = mid; else hi = mid;   // cnt is wave-uniform (SGPR)
  }
  const uint32_t T    = lo;                 // bit pattern of K-th largest |z|
  const float    tval = __uint_as_float(T);

  // ---- sum of strict-top values + tie fill at threshold ----
  int   cgt  = 0;
  float ssum = 0.0f;
  #pragma unroll
  for (int j = 0; j < ELEMS_PER_LANE; ++j) {
    if (zb[j] > T) ssum += __uint_as_float(zb[j]);
    cgt += __builtin_popcount(__builtin_amdgcn_ballot_w32(zb[j] > T));
  }
  #pragma unroll
  for (int m = 16; m >= 1; m >>= 1) ssum += __shfl_xor(ssum, m, 32);

  const float surp = (ssum + (float)(TOPK - cgt) * tval) * (1.0f / (float)TOPK);
  const float gate = 1.0f + expf(log_alpha[0]) * tanhf(expf(log_sigma[0]) * surp);

  // ---- output: exact-erf GELU * gate, re-reading x from the LDS stage ----
  #pragma unroll
  for (int c = 0; c < CHUNKS; ++c) {
    const int e = c * 128 + lane * 4;
    const float4 xv = *(const float4*)(ltile + e);
    float4 o;
    o.x = 0.5f * xv.x * (1.0f + erff(xv.x * 0.70710678118654752440f)) * gate;
    o.y = 0.5f * xv.y * (1.0f + erff(xv.y * 0.70710678118654752440f)) * gate;
    o.z = 0.5f * xv.z * (1.0f + erff(xv.z * 0.70710678118654752440f)) * gate;
    o.w = 0.5f * xv.w * (1.0f + erff(xv.w * 0.70710678118654752440f)) * gate;
    *(float4*)(orow + e) = o;
  }
}

extern "C" void kernel_launch(void* const* d_in, const int* in_sizes, int n_in,
                              void* d_out, int out_size, void* d_ws, size_t ws_size,
                              hipStream_t stream) {
  (void)n_in; (void)out_size; (void)d_ws; (void)ws_size;
  const float* x         = (const float*)d_in[0];
  const float* log_alpha = (const float*)d_in[1];
  const float* log_sigma = (const float*)d_in[2];
  const float* ema_mean  = (const float*)d_in[3];
  const float* ema_sq    = (const float*)d_in[4];
  float*       out       = (float*)d_out;

  const int n_rows = in_sizes[0] / D_DIM;                 // 8*4096 = 32768
  const int grid   = (n_rows + WAVES_PER_BLOCK - 1) / WAVES_PER_BLOCK;
  surprise_gelu_topk_kernel<<<grid, BLOCK, 0, stream>>>(
      x, log_alpha, log_sigma, ema_mean, ema_sq, out, n_rows);
}